// NCE_39822936768777
// MI455X (gfx1250) — compile-verified
//
#include <hip/hip_runtime.h>
#include <hip/hip_bf16.h>
#include <math.h>

typedef __attribute__((ext_vector_type(2))) float v2f;
typedef __attribute__((ext_vector_type(4))) float v4f;
typedef __attribute__((ext_vector_type(8))) float v8f;

#define REG_BLOCKS 2048
#define THREADS 256
#define LOSS_WAVES 8

__device__ __forceinline__ float log_sigmoid(float x) {
    // stable: min(x,0) - log1p(exp(-|x|))
    float m = fminf(x, 0.0f);
    return m - log1pf(__expf(-fabsf(x)));
}

// Block-wide sum; safe for back-to-back reuse of the same shared buffer.
__device__ __forceinline__ float block_reduce_sum(float v, float* sm) {
    int tid = threadIdx.x;
    sm[tid] = v;
    __syncthreads();
    for (int off = (int)blockDim.x >> 1; off > 0; off >>= 1) {
        if (tid < off) sm[tid] += sm[tid + off];
        __syncthreads();
    }
    float r = sm[0];
    __syncthreads();
    return r;
}

// ---------------- Fused kernel ----------------
// Blocks [0, loss_blocks)              : WMMA batched dots + NCE loss terms
// Blocks [loss_blocks, +REG_BLOCKS)    : streaming sum-of-squares over embed
// The scattered, latency-bound gathers of the loss path overlap the
// bandwidth-bound 512MB stream instead of serializing after it.
__global__ __launch_bounds__(THREADS)
void nce_fused_kernel(const float* __restrict__ embed,
                      const float* __restrict__ bias,
                      const float* __restrict__ freq,
                      const int* __restrict__ target_idx,
                      const int* __restrict__ context_idx,
                      const int* __restrict__ noise_idx,
                      int V, int B, int K, int E,
                      int loss_blocks, size_t n4,
                      float* __restrict__ loss_partial,
                      float* __restrict__ reg_partial) {
    __shared__ float cbuf[LOSS_WAVES][32][8]; // WMMA C dump per wave
    __shared__ float sarr[LOSS_WAVES][16];    // per-tile scores s[bb]
    __shared__ float red[THREADS];

    if ((int)blockIdx.x >= loss_blocks) {
        // ---------- streaming sum-of-squares (non-temporal b128 loads) ----------
        const v4f* __restrict__ emb4 = (const v4f*)embed;
        int rb = (int)blockIdx.x - loss_blocks;
        size_t tid = (size_t)rb * THREADS + threadIdx.x;
        size_t stride = (size_t)REG_BLOCKS * THREADS;
        float acc = 0.0f;
        for (size_t i = tid; i < n4; i += stride) {
            v4f v = __builtin_nontemporal_load(&emb4[i]);
            acc = fmaf(v.x, v.x, acc);
            acc = fmaf(v.y, v.y, acc);
            acc = fmaf(v.z, v.z, acc);
            acc = fmaf(v.w, v.w, acc);
        }
        float s = block_reduce_sum(acc, red);
        if (threadIdx.x == 0) reg_partial[rb] = s;
        return;
    }

    // ---------------- WMMA batched dots + NCE loss ----------------
    const int lane = threadIdx.x & 31;
    const int wave = threadIdx.x >> 5;
    const int tile = blockIdx.x * LOSS_WAVES + wave;
    const int b0   = tile * 16;           // 16 batch elements per wave
    const int t    = lane & 15;           // column (M for A, N for B) index
    const int half = lane >> 4;           // 0: K={0,1}; 1: K={2,3}

    const size_t sV = (size_t)V;
    const int ctx = context_idx[b0 + t];
    const int tgt = target_idx[b0 + t];

    // C[m,n] = sum_e r[e, b0+m] * q[e, b0+n]; diagonal = per-batch dots.
    v8f c = {0.f, 0.f, 0.f, 0.f, 0.f, 0.f, 0.f, 0.f};
    const int chunks = E >> 2; // K=4 per WMMA
    for (int ck = 0; ck < chunks; ++ck) {
        int e0 = 4 * ck + 2 * half;
        v2f a, bm;
        a.x  = embed[(size_t)e0       * sV + ctx];
        a.y  = embed[(size_t)(e0 + 1) * sV + ctx];
        bm.x = embed[(size_t)e0       * sV + tgt];
        bm.y = embed[(size_t)(e0 + 1) * sV + tgt];
        c = __builtin_amdgcn_wmma_f32_16x16x4_f32(
                /*neg_a=*/false, a, /*neg_b=*/false, bm,
                /*c_mod=*/(short)0, c, /*reuse_a=*/false, /*reuse_b=*/false);
    }

    // Dump accumulator to LDS; extract the diagonal.
#pragma unroll
    for (int i = 0; i < 8; ++i) cbuf[wave][lane][i] = c[i];
    __syncthreads();

    if (lane < 16) {
        // (t,t): t<8 -> lane t, vgpr t ; t>=8 -> lane t+16, vgpr t-8
        float dot = (lane < 8) ? cbuf[wave][lane][lane]
                               : cbuf[wave][lane + 16][lane - 8];
        sarr[wave][lane] = (dot + bias[tgt]) / (float)E;
    }
    __syncthreads();

    // Noise terms: 16 batches x K noise samples, strided over 32 lanes.
    float acc = 0.0f;
    const float fK = (float)K;
    int work = 16 * K;
    for (int i = lane; i < work; i += 32) {
        int bb = i / K;
        int k  = i - bb * K;
        int n  = noise_idx[(size_t)k * B + (b0 + bb)];
        float x = -(sarr[wave][bb] - logf(fK * freq[n]));
        acc += log_sigmoid(x);
    }
    // Target terms: one per batch element.
    if (lane < 16) {
        float x = sarr[wave][lane] - logf(fK * freq[tgt]);
        acc += log_sigmoid(x);
    }

    float ssum = block_reduce_sum(acc, red);
    if (threadIdx.x == 0) loss_partial[blockIdx.x] = ssum;
}

// ---------------- Finalize: fold partials into the scalar loss ----------------
__global__ __launch_bounds__(256)
void finalize_kernel(const float* __restrict__ reg_partial, int nreg,
                     const float* __restrict__ loss_partial, int nloss,
                     int V, int B, int E, float* __restrict__ out) {
    __shared__ float sm[256];
    float a = 0.0f;
    for (int i = threadIdx.x; i < nreg; i += (int)blockDim.x) a += reg_partial[i];
    float sumsq = block_reduce_sum(a, sm);

    float l = 0.0f;
    for (int i = threadIdx.x; i < nloss; i += (int)blockDim.x) l += loss_partial[i];
    float loss = block_reduce_sum(l, sm);

    if (threadIdx.x == 0) {
        float reg = 0.1f * sumsq / ((float)E * (float)V);
        out[0] = -loss / (float)B + reg;
    }
}

extern "C" void kernel_launch(void* const* d_in, const int* in_sizes, int n_in,
                              void* d_out, int out_size, void* d_ws, size_t ws_size,
                              hipStream_t stream) {
    (void)n_in; (void)out_size; (void)ws_size;
    const float* embed       = (const float*)d_in[0];
    const float* bias        = (const float*)d_in[1];
    const float* freq        = (const float*)d_in[2];
    const int*   target_idx  = (const int*)d_in[3];
    const int*   context_idx = (const int*)d_in[4];
    const int*   noise_idx   = (const int*)d_in[5];

    const int V = in_sizes[2];          // vocab
    const int B = in_sizes[3];          // batch (4096)
    const int K = in_sizes[5] / B;      // noise count (64)
    const int E = in_sizes[0] / V;      // embed dim (128)

    float* ws = (float*)d_ws;
    float* reg_partial  = ws;                 // REG_BLOCKS floats
    float* loss_partial = ws + REG_BLOCKS;    // loss_blocks floats

    const int tiles = B / 16;
    const int loss_blocks = tiles / LOSS_WAVES;

    size_t n4 = ((size_t)E * (size_t)V) / 4;

    nce_fused_kernel<<<loss_blocks + REG_BLOCKS, THREADS, 0, stream>>>(
        embed, bias, freq, target_idx, context_idx, noise_idx,
        V, B, K, E, loss_blocks, n4, loss_partial, reg_partial);

    finalize_kernel<<<1, 256, 0, stream>>>(
        reg_partial, REG_BLOCKS, loss_partial, loss_blocks,
        V, B, E, (float*)d_out);
}